// PPD_39058432590486
// MI455X (gfx1250) — compile-verified
//
#include <hip/hip_runtime.h>

// ---------------------------------------------------------------------------
// Masked gather + mean((1 - logits[row, target])^2) over rows with target!=-1
//
// Memory/latency-bound: ~264 MB effective traffic (8 MB coalesced int32
// targets + ~1 cacheline per gathered row) -> ~11us at 23.3 TB/s HBM.
// Strategy:
//   * 2x int4 target loads per thread (coalesced, non-temporal)
//   * 8 UNCONDITIONAL clamped-index gathers per thread (branchless; all 8
//     in flight -> one loadcnt wait), contributions masked with selects.
//     NT hints everywhere: 671 MB footprint > 192 MB L2, zero reuse.
//   * cross-lane reduction of (sum, count) done with a single
//     V_WMMA_F32_16X16X4_F32 per wave (dual-column ones-matrix trick),
//     then deterministic 2-pass block reduction through d_ws.
// ---------------------------------------------------------------------------

typedef __attribute__((ext_vector_type(2))) float v2f;
typedef __attribute__((ext_vector_type(8))) float v8f;
typedef __attribute__((ext_vector_type(4))) int   v4i;

#define NUM_CLASSES 80
#define ROWS_PER_THREAD 8

// Reduce BOTH sv and cv across all 32 lanes of the wave with one
// V_WMMA_F32_16X16X4_F32 (D = A x B, f32 throughout, deterministic).
//
// A (16x4 f32) layout: lane m (0..15): VGPR0=K0, VGPR1=K1; lane m+16: K2,K3.
//   -> a.x = sv, a.y = cv gives A[m][0]=sv_m, A[m][1]=cv_m,
//                              A[m][2]=sv_{m+16}, A[m][3]=cv_{m+16}
// B (4x16 f32) layout: VGPR0 = rows {0,2} (lane halves), VGPR1 = rows {1,3}.
//   Choose B[k][n] = (k even) ? (n<8) : (n>=8), so
//     D[m][n<8]  = sv_m + sv_{m+16}   (sum partials)
//     D[m][n>=8] = cv_m + cv_{m+16}   (count partials)
// D layout: VGPR v, lanes 0-15 -> M=v, lanes 16-31 -> M=v+8, N=lane%16.
// t = sum(d[0..7]); t += shfl_xor(t,16): lane 0 holds total sum,
// lane 8 holds total count.
__device__ __forceinline__ void wave_reduce2_wmma(float sv, float cv,
                                                  float& s_out, float& c_out) {
  v2f a;
  a.x = sv;
  a.y = cv;
  const int n = threadIdx.x & 15;
  v2f b;
  b.x = (n < 8) ? 1.0f : 0.0f;  // rows K=0 / K=2 -> sum columns 0..7
  b.y = (n < 8) ? 0.0f : 1.0f;  // rows K=1 / K=3 -> count columns 8..15
  v8f c = {};
  v8f d = __builtin_amdgcn_wmma_f32_16x16x4_f32(
      /*neg_a=*/false, a, /*neg_b=*/false, b,
      /*c_mod=*/(short)0, c, /*reuse_a=*/false, /*reuse_b=*/false);
  float t = d[0] + d[1] + d[2] + d[3] + d[4] + d[5] + d[6] + d[7];
  t += __shfl_xor(t, 16, 32);
  s_out = __shfl(t, 0, 32);  // column 0  -> total sum
  c_out = __shfl(t, 8, 32);  // column 8  -> total count
}

// Kernel 1: each thread handles 8 consecutive rows branchlessly; per-block
// (sum,count) partial written to d_ws. 256 threads = 8 wave32s per block.
__global__ void __launch_bounds__(256) ppd_partial_kernel(
    const float* __restrict__ logits, const int* __restrict__ target,
    float2* __restrict__ partials, int n_rows) {
  const int tid = blockIdx.x * blockDim.x + threadIdx.x;

  float s = 0.0f, c = 0.0f;
  const unsigned row0 = (unsigned)tid * ROWS_PER_THREAD;
  if (row0 + (ROWS_PER_THREAD - 1) < (unsigned)n_rows) {
    // Coalesced, non-temporal 32B of targets per lane (wave reads 1KB).
    const v4i ta = __builtin_nontemporal_load((const v4i*)target + 2 * tid);
    const v4i tb = __builtin_nontemporal_load((const v4i*)target + 2 * tid + 1);
    int t[ROWS_PER_THREAD];
#pragma unroll
    for (int j = 0; j < 4; ++j) {
      t[j] = ta[j];
      t[4 + j] = tb[j];
    }

    // Issue all 8 gathers unconditionally (clamped index) so they are all
    // in flight before a single loadcnt wait. 32-bit offsets keep the
    // addressing in SGPR-base + voffset form.
    float x[ROWS_PER_THREAD];
#pragma unroll
    for (int j = 0; j < ROWS_PER_THREAD; ++j) {
      const unsigned idx =
          (row0 + (unsigned)j) * NUM_CLASSES + (unsigned)(t[j] < 0 ? 0 : t[j]);
      x[j] = __builtin_nontemporal_load(logits + idx);
    }

    // Branchless masked accumulation.
#pragma unroll
    for (int j = 0; j < ROWS_PER_THREAD; ++j) {
      const bool m = (t[j] >= 0);  // IGNORE_LABEL == -1
      const float dlt = m ? (1.0f - x[j]) : 0.0f;
      s = fmaf(dlt, dlt, s);
      c += m ? 1.0f : 0.0f;
    }
  }

  // All threads reconverge here -> EXEC all ones, as WMMA requires.
  float ws_s, ws_c;
  wave_reduce2_wmma(s, c, ws_s, ws_c);

  __shared__ float2 wacc[8];
  const int wave = threadIdx.x >> 5;
  if ((threadIdx.x & 31) == 0) wacc[wave] = make_float2(ws_s, ws_c);
  __syncthreads();
  if (threadIdx.x == 0) {
    float bs = 0.0f, bc = 0.0f;
#pragma unroll
    for (int w = 0; w < 8; ++w) {
      bs += wacc[w].x;
      bc += wacc[w].y;
    }
    partials[blockIdx.x] = make_float2(bs, bc);  // fixed order: deterministic
  }
}

// Kernel 2: single block reduces all per-block partials, writes sum/count.
__global__ void __launch_bounds__(256) ppd_final_kernel(
    const float2* __restrict__ partials, int nparts, float* __restrict__ out) {
  float s = 0.0f, c = 0.0f;
  for (int i = threadIdx.x; i < nparts; i += blockDim.x) {
    const float2 p = partials[i];
    s += p.x;
    c += p.y;
  }

  float ws_s, ws_c;
  wave_reduce2_wmma(s, c, ws_s, ws_c);

  __shared__ float2 wacc[8];
  const int wave = threadIdx.x >> 5;
  if ((threadIdx.x & 31) == 0) wacc[wave] = make_float2(ws_s, ws_c);
  __syncthreads();
  if (threadIdx.x == 0) {
    float bs = 0.0f, bc = 0.0f;
#pragma unroll
    for (int w = 0; w < 8; ++w) {
      bs += wacc[w].x;
      bc += wacc[w].y;
    }
    out[0] = bs / bc;
  }
}

extern "C" void kernel_launch(void* const* d_in, const int* in_sizes, int n_in,
                              void* d_out, int out_size, void* d_ws,
                              size_t ws_size, hipStream_t stream) {
  const float* logits = (const float*)d_in[0];  // [N, 80] f32
  const int* target = (const int*)d_in[1];      // [N] int
  const int n_rows = in_sizes[1];               // N = 2097152

  const int threads = 256;
  const int rows_per_block = threads * ROWS_PER_THREAD;  // 2048 rows / block
  const int nblocks = (n_rows + rows_per_block - 1) / rows_per_block;  // 1024

  float2* partials = (float2*)d_ws;  // nblocks * 8 bytes = 8 KB scratch

  ppd_partial_kernel<<<nblocks, threads, 0, stream>>>(logits, target, partials,
                                                      n_rows);
  ppd_final_kernel<<<1, threads, 0, stream>>>(partials, nblocks,
                                              (float*)d_out);
}